// ForWard_27891517621100
// MI455X (gfx1250) — compile-verified
//
#include <hip/hip_runtime.h>

typedef __attribute__((ext_vector_type(2))) float v2f;
typedef __attribute__((ext_vector_type(8))) float v8f;

constexpr int BATCH   = 16384;
constexpr int WIDTH   = 128;   // layer width
constexpr int KP      = 129;   // preds + bias row
constexpr int NWL     = 31;    // weighted layers
constexpr int WSTRIDE = 132;   // padded k-stride of transposed masked weights
constexpr int ASTRIDE = 132;   // padded col-stride of activation tile in LDS
constexpr int ROWS_PER_WG = 128;
constexpr int THREADS = 256;   // 8 wave32s, tiled 2 (rows) x 4 (cols)

// ---------------------------------------------------------------------------
// Kernel 1: fuse mask*weight and transpose to [l][j][k] with padded k-stride.
// weights/masks are [NWL][KP][WIDTH] row-major f32.
// ---------------------------------------------------------------------------
__global__ void prep_weights(const float* __restrict__ w,
                             const float* __restrict__ m,
                             float* __restrict__ wt) {
    int idx = blockIdx.x * blockDim.x + threadIdx.x;
    int total = NWL * WIDTH * KP;
    if (idx >= total) return;
    int k = idx % KP;
    int j = (idx / KP) % WIDTH;
    int l = idx / (KP * WIDTH);
    int src = (l * KP + k) * WIDTH + j;
    wt[((size_t)l * WIDTH + j) * WSTRIDE + k] = w[src] * m[src];
}

// ---------------------------------------------------------------------------
// Kernel 2: 31-layer MLP via WMMA f32 16x16x4; activations resident in LDS.
// Block tile: 128 rows x 128 cols. Wave tile: 64 rows x 32 cols
// (acc[4][2] of v8f = 64 VGPRs; 4 A-frags + 2 B-frags per K-step -> 8 WMMA).
// ---------------------------------------------------------------------------
__global__ void __launch_bounds__(THREADS, 1)
dag_mlp(const float* __restrict__ x,
        const float* __restrict__ wt,    // [NWL][WIDTH][WSTRIDE] transposed masked weights
        float* __restrict__ out) {
    __shared__ __align__(16) float sW[WIDTH * WSTRIDE];        // 67584 B
    __shared__ __align__(16) float sA[ROWS_PER_WG * ASTRIDE];  // 67584 B

    const int tid   = threadIdx.x;
    const int lane  = tid & 31;
    const int wave  = tid >> 5;       // 0..7
    const int m     = lane & 15;
    const int hi    = lane >> 4;
    const int rowB  = (wave >> 2) * 64;   // wave row base   (0 or 64)
    const int colB  = (wave & 3) * 32;    // wave col base   (0,32,64,96)
    const size_t gRow0 = (size_t)blockIdx.x * ROWS_PER_WG;

    // ---- load x tile into sA (row-major, padded stride) ----
    {
        const float4* xv = reinterpret_cast<const float4*>(x + gRow0 * WIDTH);
        for (int i = tid; i < ROWS_PER_WG * WIDTH / 4; i += THREADS) {
            int r  = i >> 5;   // 32 float4 per 128-wide row
            int c4 = i & 31;
            *reinterpret_cast<float4*>(&sA[r * ASTRIDE + c4 * 4]) = xv[r * 32 + c4];
        }
    }
    __syncthreads();

    for (int l = 0; l < NWL; ++l) {
        // ---- stage this layer's masked weights into LDS (straight copy) ----
        const float4* wv = reinterpret_cast<const float4*>(wt + (size_t)l * WIDTH * WSTRIDE);
        for (int i = tid; i < WIDTH * WSTRIDE / 4; i += THREADS) {
            *reinterpret_cast<float4*>(&sW[i * 4]) = wv[i];
        }
        __syncthreads();

        // ---- warm L2/WGP$ for the next layer's weights during compute ----
        if (l + 1 < NWL) {
            const char* nxt = reinterpret_cast<const char*>(
                wt + (size_t)(l + 1) * WIDTH * WSTRIDE);
            __builtin_prefetch(nxt + tid * 264, 0, 3);
        }

        // ---- init accumulators with bias (k = 128 row of transposed weights) ----
        v8f acc[4][2];
#pragma unroll
        for (int nt = 0; nt < 2; ++nt) {
            float bval = sW[(colB + nt * 16 + m) * WSTRIDE + 128];
#pragma unroll
            for (int mt = 0; mt < 4; ++mt)
#pragma unroll
                for (int r = 0; r < 8; ++r) acc[mt][nt][r] = bval;
        }

        // ---- K loop: 32 steps of K=4 ----
#pragma unroll 4
        for (int kk = 0; kk < 32; ++kk) {
            const int kb = kk * 4 + hi * 2;
            v2f a[4], b[2];
#pragma unroll
            for (int mt = 0; mt < 4; ++mt)
                a[mt] = *reinterpret_cast<const v2f*>(
                    &sA[(rowB + mt * 16 + m) * ASTRIDE + kb]);
#pragma unroll
            for (int nt = 0; nt < 2; ++nt)
                b[nt] = *reinterpret_cast<const v2f*>(
                    &sW[(colB + nt * 16 + m) * WSTRIDE + kb]);
#pragma unroll
            for (int mt = 0; mt < 4; ++mt)
#pragma unroll
                for (int nt = 0; nt < 2; ++nt)
                    acc[mt][nt] = __builtin_amdgcn_wmma_f32_16x16x4_f32(
                        false, a[mt], false, b[nt], (short)0, acc[mt][nt],
                        false, false);
        }

        // ---- ReLU + write wave's 64x32 tile back to sA (C layout) ----
#pragma unroll
        for (int mt = 0; mt < 4; ++mt)
#pragma unroll
            for (int nt = 0; nt < 2; ++nt)
#pragma unroll
                for (int r = 0; r < 8; ++r) {
                    float v = fmaxf(acc[mt][nt][r], 0.0f);
                    sA[(rowB + mt * 16 + r + 8 * hi) * ASTRIDE +
                       colB + nt * 16 + m] = v;
                }
        __syncthreads();   // acts complete + sW safe to overwrite
    }

    // ---- write final activations to d_out [BATCH][WIDTH] ----
    {
        float4* ov = reinterpret_cast<float4*>(out + gRow0 * WIDTH);
        for (int i = tid; i < ROWS_PER_WG * WIDTH / 4; i += THREADS) {
            int r  = i >> 5;
            int c4 = i & 31;
            ov[r * 32 + c4] = *reinterpret_cast<const float4*>(&sA[r * ASTRIDE + c4 * 4]);
        }
    }
}

// ---------------------------------------------------------------------------
extern "C" void kernel_launch(void* const* d_in, const int* in_sizes, int n_in,
                              void* d_out, int out_size, void* d_ws, size_t ws_size,
                              hipStream_t stream) {
    const float* x   = (const float*)d_in[0];   // [16384][128]
    const float* w   = (const float*)d_in[1];   // [31][129][128]
    const float* msk = (const float*)d_in[2];   // [31][129][128]
    float* wt = (float*)d_ws;                   // [31][128][132] ~ 2.1 MB

    int total = NWL * WIDTH * KP;
    prep_weights<<<(total + 255) / 256, 256, 0, stream>>>(w, msk, wt);

    dag_mlp<<<BATCH / ROWS_PER_WG, THREADS, 0, stream>>>(x, wt, (float*)d_out);
}